// EffectiveResistanceEdgeMask_45921790329384
// MI455X (gfx1250) — compile-verified
//
#include <hip/hip_runtime.h>
#include <hip/hip_bf16.h>

// ---------------- problem constants (fixed by reference setup) ----------------
#define N_NODES 4096
#define M_EDGES 8192
#define NS_ITERS 20          // Newton-Schulz iterations for SPD inverse
#define REG 1e-3f

typedef __attribute__((ext_vector_type(2))) float v2f;
typedef __attribute__((ext_vector_type(4))) float v4f;
typedef __attribute__((ext_vector_type(8))) float v8f;

#define AS1 __attribute__((address_space(1)))
#define AS3 __attribute__((address_space(3)))

// pointee type must exactly match the builtin's declared param: vector_size(16) int
typedef int v4i_vs __attribute__((__vector_size__(16)));

// ---- CDNA5 async direct global->LDS copy (16 bytes per lane), ASYNCcnt-tracked --
__device__ __forceinline__ void async_copy16(const float* g, float* l) {
#if __has_builtin(__builtin_amdgcn_global_load_async_to_lds_b128)
    __builtin_amdgcn_global_load_async_to_lds_b128(
        (AS1 v4i_vs*)(unsigned long long)g,
        (AS3 v4i_vs*)(unsigned)(unsigned long long)l, 0, 0);
#else
    unsigned lds_off = (unsigned)(unsigned long long)l;  // addr[31:0] == LDS offset
    asm volatile("global_load_async_to_lds_b128 %0, %1, off"
                 :: "v"(lds_off), "v"(g) : "memory");
#endif
}

__device__ __forceinline__ void wait_async0() {
#if __has_builtin(__builtin_amdgcn_s_wait_asynccnt)
    __builtin_amdgcn_s_wait_asynccnt(0);
#else
    asm volatile("s_wait_asynccnt 0x0" ::: "memory");
#endif
}

// ---------------- 1) build incidence bitmask (column-major: per-edge bit rows) ---
__global__ void build_bits_kernel(const int* __restrict__ V, const int* __restrict__ E,
                                  unsigned* __restrict__ Hbits, int nnz, int wpe) {
    int i = blockIdx.x * blockDim.x + threadIdx.x;
    if (i >= nnz) return;
    int v = V[i], e = E[i];
    atomicOr(&Hbits[(size_t)e * wpe + (v >> 5)], 1u << (v & 31));
}

// ---------------- 2) per-edge scan + Laplacian scatter  L -= de_inv over pairs ---
#define MAXMEM 256
__global__ __launch_bounds__(128)
void edge_scan_scatter_kernel(const unsigned* __restrict__ Hbits,
                              float* __restrict__ Lmat, float* __restrict__ dv, int n) {
    __shared__ int list[MAXMEM];
    __shared__ int tmp[128];
    const int e = blockIdx.x;
    const int t = threadIdx.x;                    // one 32-bit word per thread
    unsigned w = Hbits[(size_t)e * (n / 32) + t];
    int cnt = __popc(w);
    tmp[t] = cnt;
    __syncthreads();
    // Hillis-Steele inclusive scan (deterministic member ordering)
    for (int off = 1; off < 128; off <<= 1) {
        int val = (t >= off) ? tmp[t - off] : 0;
        __syncthreads();
        tmp[t] += val;
        __syncthreads();
    }
    int total = tmp[127];
    int pos = tmp[t] - cnt;
    unsigned ww = w;
    while (ww) {
        int b = __ffs(ww) - 1;
        ww &= ww - 1;
        if (pos < MAXMEM) list[pos] = t * 32 + b;
        pos++;
    }
    __syncthreads();
    if (total == 0) return;                       // uniform exit
    int de = total > MAXMEM ? MAXMEM : total;
    float de_inv = 1.0f / ((float)total + 1e-8f);
    for (int i = t; i < de; i += 128) atomicAdd(&dv[list[i]], 1.0f);
    int tot = de * de;                            // includes diagonal (u,u) pairs
    for (int p = t; p < tot; p += 128) {
        int u = list[p / de];
        int v = list[p % de];
        atomicAdd(&Lmat[(size_t)u * n + v], -de_inv);
    }
}

// ---------------- 3) A = L + Dv + reg*I on the diagonal --------------------------
__global__ void add_diag_kernel(float* __restrict__ A, const float* __restrict__ dv,
                                int n, float reg) {
    int i = blockIdx.x * blockDim.x + threadIdx.x;
    if (i < n) A[(size_t)i * n + i] += dv[i] + reg;
}

// ---------------- 4) ||A||_inf = max row abs-sum (A symmetric => == ||A||_1) -----
__global__ __launch_bounds__(256)
void row_abs_max_kernel(const float* __restrict__ A, unsigned* __restrict__ rmax_bits,
                        int n) {
    __shared__ float red[256];
    int row = blockIdx.x;
    float s = 0.0f;
    for (int c = threadIdx.x; c < n; c += 256) s += fabsf(A[(size_t)row * n + c]);
    red[threadIdx.x] = s;
    __syncthreads();
    for (int off = 128; off > 0; off >>= 1) {
        if (threadIdx.x < off) red[threadIdx.x] += red[threadIdx.x + off];
        __syncthreads();
    }
    if (threadIdx.x == 0) atomicMax(rmax_bits, __float_as_uint(red[0]));
}

// ---------------- 5) X0 = A / (||A||_1 * ||A||_inf) ------------------------------
__global__ void scale_init_kernel(float* __restrict__ X, const float* __restrict__ A,
                                  const unsigned* __restrict__ rmax_bits, size_t nn) {
    size_t i = (size_t)blockIdx.x * blockDim.x + threadIdx.x;
    if (i >= nn) return;
    float r = __uint_as_float(*rmax_bits);
    float alpha = 1.0f / (r * r + 1e-30f);
    X[i] = A[i] * alpha;
}

// ---------------- 6) W = 2I - Y (in place) ---------------------------------------
__global__ void neg_plus_2i_kernel(float* __restrict__ Y, int n) {
    size_t i = (size_t)blockIdx.x * blockDim.x + threadIdx.x;
    size_t nn = (size_t)n * n;
    if (i >= nn) return;
    float w = -Y[i];
    if ((i / (size_t)n) == (i % (size_t)n)) w += 2.0f;
    Y[i] = w;
}

// ---------------- 7) fp32 WMMA GEMM: C = A * B  (n x n, row-major, n%128==0) -----
// 128x128 block tile, 8 waves of 64x32, double-buffered LDS fed by async
// direct-to-LDS copies (ASYNCcnt) so tile k+1 streams in under tile k's WMMAs.
#define BM 128
#define BN 128
#define BK 32
#define LDA 36      // padded LDS stride for A tile (rows stay 16B-addressable)

__global__ __launch_bounds__(256)
void gemm_f32_wmma_kernel(float* __restrict__ C, const float* __restrict__ A,
                          const float* __restrict__ B, int n) {
    __shared__ float As[2][BM * LDA];   // [buf][row][k]
    __shared__ float Bs[2][BK * BN];    // [buf][k][col]

    const int tid  = threadIdx.x;
    const int lane = tid & 31;       // wave32
    const int wave = tid >> 5;       // 0..7
    const int l15  = lane & 15;
    const int lhi  = lane >> 4;      // 0/1 -> K-half selector per ISA 16x4 layout
    const int wm0  = (wave & 1) * 64;
    const int wn0  = (wave >> 1) * 32;
    const int bm   = blockIdx.y * BM;
    const int bn   = blockIdx.x * BN;

    v8f acc[4][2] = {};              // 4x2 grid of 16x16 f32 accumulators

    const int arow  = tid >> 3;           // 0..31
    const int acol4 = (tid & 7) * 4;      // float4 column within K-tile
    const int brow  = tid >> 5;           // 0..7
    const int bcol4 = (tid & 31) * 4;     // float4 column within N-tile

    auto issue_tile = [&](int buf, int k0) {
        #pragma unroll
        for (int p = 0; p < 4; p++) {     // A tile: 128 x 32, row-contiguous 16B
            int r = arow + p * 32;
            async_copy16(&A[(size_t)(bm + r) * n + k0 + acol4],
                         &As[buf][r * LDA + acol4]);
        }
        #pragma unroll
        for (int p = 0; p < 4; p++) {     // B tile: 32 x 128, row-contiguous 16B
            int r = brow + p * 8;
            async_copy16(&B[(size_t)(k0 + r) * n + bn + bcol4],
                         &Bs[buf][r * BN + bcol4]);
        }
    };

    issue_tile(0, 0);
    const int ntiles = n / BK;
    for (int kt = 0; kt < ntiles; ++kt) {
        const int cur = kt & 1;
        wait_async0();                    // my async writes into buf[cur] done
        __syncthreads();                  // everyone's writes visible
        if (kt + 1 < ntiles) issue_tile(cur ^ 1, (kt + 1) * BK);  // prefetch next

        const float* __restrict__ as = &As[cur][0];
        const float* __restrict__ bs = &Bs[cur][0];
        #pragma unroll
        for (int kk = 0; kk < BK; kk += 4) {
            const int ka = kk + 2 * lhi;  // K pair selected by lane-half
            v2f af[4];
            #pragma unroll
            for (int i = 0; i < 4; i++)
                af[i] = *(const v2f*)&as[(wm0 + i * 16 + l15) * LDA + ka];
            v2f bf[2];
            #pragma unroll
            for (int j = 0; j < 2; j++) {
                v2f b;
                b.x = bs[ka * BN + wn0 + j * 16 + l15];
                b.y = bs[(ka + 1) * BN + wn0 + j * 16 + l15];
                bf[j] = b;
            }
            #pragma unroll
            for (int i = 0; i < 4; i++)
                #pragma unroll
                for (int j = 0; j < 2; j++)
                    acc[i][j] = __builtin_amdgcn_wmma_f32_16x16x4_f32(
                        false, af[i], false, bf[j], (short)0, acc[i][j], false, false);
        }
        __syncthreads();                  // all reads of buf[cur] consumed
    }
    // C/D layout: lane hi-half holds rows M+8.., VGPR r -> row offset r
    #pragma unroll
    for (int i = 0; i < 4; i++)
        #pragma unroll
        for (int j = 0; j < 2; j++) {
            int col = bn + wn0 + j * 16 + l15;
            #pragma unroll
            for (int r = 0; r < 8; r++) {
                int row = bm + wm0 + i * 16 + r + 8 * lhi;
                C[(size_t)row * n + col] = acc[i][j][r];
            }
        }
}

// ---------------- 8) tail kernels -------------------------------------------------
__global__ void extract_diag_kernel(const float* __restrict__ X, float* __restrict__ dg,
                                    int n) {
    int i = blockIdx.x * blockDim.x + threadIdx.x;
    if (i < n) dg[i] = X[(size_t)i * (n + 1)];
}

__global__ void pair_scatter_kernel(const int* __restrict__ V, const int* __restrict__ E,
                                    const float* __restrict__ dg, float* __restrict__ esum,
                                    int* __restrict__ ecnt, int nnz) {
    int i = blockIdx.x * blockDim.x + threadIdx.x;
    if (i >= nnz) return;
    atomicAdd(&esum[E[i]], dg[V[i]]);
    atomicAdd(&ecnt[E[i]], 1);
}

__global__ void compute_probs_kernel(const float* __restrict__ esum,
                                     const int* __restrict__ ecnt,
                                     float* __restrict__ probs_ws,
                                     float* __restrict__ out, int m) {
    int e = blockIdx.x * blockDim.x + threadIdx.x;
    if (e >= m) return;
    float p = (ecnt[e] > 1) ? esum[e] : 0.0f;
    probs_ws[e] = p;
    out[e] = p;
}

__global__ __launch_bounds__(256)
void topk_mask_kernel(const float* __restrict__ probs, float* __restrict__ out_soft,
                      float* __restrict__ out_hard, int m, int k) {
    __shared__ float sp[M_EDGES];    // 32 KB of the 320 KB LDS
    for (int i = threadIdx.x; i < m; i += blockDim.x) sp[i] = probs[i];
    __syncthreads();
    int e = blockIdx.x * blockDim.x + threadIdx.x;
    if (e >= m) return;
    float p = sp[e];
    int rank = 0;
    for (int j = 0; j < m; j++) {
        float pj = sp[j];
        rank += (pj > p) || (pj == p && j < e);   // stable tie-break like top_k
    }
    float hard = (rank < k) ? 1.0f : 0.0f;
    out_hard[e] = hard;
    out_soft[e] = (hard - p) + p;                 // straight-through as written
}

// ---------------- launcher --------------------------------------------------------
extern "C" void kernel_launch(void* const* d_in, const int* in_sizes, int n_in,
                              void* d_out, int out_size, void* d_ws, size_t ws_size,
                              hipStream_t stream) {
    const int n = N_NODES, m = M_EDGES;
    const int nnz = in_sizes[1];
    const int* V = (const int*)d_in[1];
    const int* E = (const int*)d_in[2];
    float* out = (float*)d_out;

    char* ws = (char*)d_ws;
    size_t off = 0;
    auto carve = [&](size_t bytes) -> void* {
        void* p = ws + off;
        off += (bytes + 255) & ~(size_t)255;
        return p;
    };
    const size_t nnB = (size_t)n * n * sizeof(float);
    unsigned* Hbits = (unsigned*)carve((size_t)m * (n / 32) * sizeof(unsigned));
    float* Amat = (float*)carve(nnB);
    float* X    = (float*)carve(nnB);
    float* Y    = (float*)carve(nnB);
    float* Z    = (float*)carve(nnB);
    float* dv   = (float*)carve(n * sizeof(float));
    unsigned* rmaxb = (unsigned*)carve(256);
    float* dg   = (float*)carve(n * sizeof(float));
    float* esum = (float*)carve(m * sizeof(float));
    int*   ecnt = (int*)carve(m * sizeof(int));
    float* probsb = (float*)carve(m * sizeof(float));
    (void)ws_size; (void)n_in; (void)out_size;

    (void)hipMemsetAsync(Hbits, 0, (size_t)m * (n / 32) * sizeof(unsigned), stream);
    (void)hipMemsetAsync(Amat, 0, nnB, stream);
    (void)hipMemsetAsync(dv, 0, n * sizeof(float), stream);
    (void)hipMemsetAsync(rmaxb, 0, sizeof(unsigned), stream);
    (void)hipMemsetAsync(esum, 0, m * sizeof(float), stream);
    (void)hipMemsetAsync(ecnt, 0, m * sizeof(int), stream);

    build_bits_kernel<<<(nnz + 255) / 256, 256, 0, stream>>>(V, E, Hbits, nnz, n / 32);
    edge_scan_scatter_kernel<<<m, 128, 0, stream>>>(Hbits, Amat, dv, n);
    add_diag_kernel<<<(n + 255) / 256, 256, 0, stream>>>(Amat, dv, n, REG);
    row_abs_max_kernel<<<n, 256, 0, stream>>>(Amat, rmaxb, n);

    const size_t nn = (size_t)n * n;
    const int nnBlocks = (int)((nn + 255) / 256);
    scale_init_kernel<<<nnBlocks, 256, 0, stream>>>(X, Amat, rmaxb, nn);

    dim3 ggrid(n / BN, n / BM);
    for (int it = 0; it < NS_ITERS; ++it) {
        gemm_f32_wmma_kernel<<<ggrid, 256, 0, stream>>>(Y, Amat, X, n);   // Y = A*X
        neg_plus_2i_kernel<<<nnBlocks, 256, 0, stream>>>(Y, n);           // Y = 2I - Y
        gemm_f32_wmma_kernel<<<ggrid, 256, 0, stream>>>(Z, X, Y, n);      // Z = X*(2I-AX)
        float* t = X; X = Z; Z = t;
    }

    extract_diag_kernel<<<(n + 255) / 256, 256, 0, stream>>>(X, dg, n);
    pair_scatter_kernel<<<(nnz + 255) / 256, 256, 0, stream>>>(V, E, dg, esum, ecnt, nnz);
    compute_probs_kernel<<<(m + 255) / 256, 256, 0, stream>>>(esum, ecnt, probsb, out, m);
    topk_mask_kernel<<<(m + 255) / 256, 256, 0, stream>>>(probsb, out + m, out + 2 * m,
                                                          m, m / 2);
}